// BaseQuantizer_40518721470847
// MI455X (gfx1250) — compile-verified
//
#include <hip/hip_runtime.h>

typedef __attribute__((ext_vector_type(2))) float v2f;
typedef __attribute__((ext_vector_type(8))) float v8f;

#define D_DIM    512
#define D_STEPS  128          // D_DIM / 4 (K per WMMA = 4)
#define ROW_PAD  516          // LDS row stride in floats: 4*r bank advance -> conflict-free
#define N_CHUNK  16
#define THREADS  128          // 4 waves, each owns 16 rows (M)
#define M_TILE   64           // rows per block

// --------------------------------------------------------------------------
// Kernel 1: cnorm[k] = ||codebook[k]||^2
// --------------------------------------------------------------------------
__global__ void vq_cnorm_kernel(const float* __restrict__ cb,
                                float* __restrict__ cnorm, int K) {
    int k = blockIdx.x * blockDim.x + threadIdx.x;
    if (k >= K) return;
    const float4* row = (const float4*)(cb + (size_t)k * D_DIM);
    float s = 0.0f;
#pragma unroll 8
    for (int i = 0; i < D_DIM / 4; ++i) {
        float4 v = row[i];
        s += v.x * v.x + v.y * v.y + v.z * v.z + v.w * v.w;
    }
    cnorm[k] = s;
}

// --------------------------------------------------------------------------
// Kernel 2: per 64-row block, scan all K codes with fp32 WMMA, track argmin
// of (||w||^2 - 2 z.w). Codebook chunks are double-buffered in LDS and
// staged with CDNA5 GLOBAL_LOAD_ASYNC_TO_LDS_B128 (ASYNCcnt), overlapping
// the global->LDS copy with the 128-WMMA compute on the other buffer.
// --------------------------------------------------------------------------
__launch_bounds__(THREADS)
__global__ void vq_argmin_kernel(const float* __restrict__ z,
                                 const float* __restrict__ cb,
                                 const float* __restrict__ cnorm,
                                 float* __restrict__ zq,
                                 int* __restrict__ idx_out,
                                 int K) {
    __shared__ float ldsB[2][N_CHUNK * ROW_PAD];  // 2 x 33,024 B codebook chunks
    __shared__ int   bestIdxSh[M_TILE];

    const int tid     = threadIdx.x;
    const int lane    = tid & 31;
    const int wave    = tid >> 5;             // 0..3, each wave owns 16 rows
    const int nl      = lane & 15;            // N / M sub-index inside wave
    const int hi      = (lane >> 4) & 1;      // selects K-pair {0,1} vs {2,3}
    const int rowBase = blockIdx.x * M_TILE;

    // Async stage of one 16x512 f32 codebook chunk into LDS buffer `buf`.
    // 2048 B128 transfers, 16 per thread -> ASYNCcnt +16 per wave.
    auto stage = [&](int nBase, int buf) {
        uint32_t lds0 = (uint32_t)(uintptr_t)&ldsB[buf][0];
#pragma unroll
        for (int kk = 0; kk < 16; ++kk) {
            int i  = tid + kk * THREADS;      // 0..2047
            int r  = i >> 7;                  // code row 0..15
            int c4 = i & 127;                 // float4 column
            uint32_t goff  = ((uint32_t)(nBase + r) * D_DIM + 4u * c4) * 4u;
            uint32_t laddr = lds0 + ((uint32_t)r * ROW_PAD + 4u * c4) * 4u;
            asm volatile("global_load_async_to_lds_b128 %0, %1, %2"
                         :: "v"(laddr), "v"(goff), "s"(cb) : "memory");
        }
    };

    // ---- A fragment (16 rows x 512) resident in registers: 128 x v2f ----
    // A layout (16x4 f32 WMMA): lanes 0-15 -> M=lane, K={0,1}; lanes 16-31 -> M=lane-16, K={2,3}
    v2f afrag[D_STEPS];
    {
        const float* aptr = z + (size_t)(rowBase + wave * 16 + nl) * D_DIM + 2 * hi;
#pragma unroll
        for (int j = 0; j < D_STEPS; ++j)
            afrag[j] = *(const v2f*)(aptr + 4 * j);
    }

    float bestVal[8];
    int   bestIdx[8];
#pragma unroll
    for (int v = 0; v < 8; ++v) { bestVal[v] = 3.402823466e+38f; bestIdx[v] = 0; }

    const int nChunks = K / N_CHUNK;          // 512
    stage(0, 0);                              // prologue: fill buffer 0

    for (int c = 0; c < nChunks; ++c) {
        const int nBase = c * N_CHUNK;
        if (c + 1 < nChunks) {
            stage(nBase + N_CHUNK, (c + 1) & 1);          // prefetch next chunk
            asm volatile("s_wait_asynccnt 0x10" ::: "memory");  // chunk c landed
        } else {
            asm volatile("s_wait_asynccnt 0x0" ::: "memory");
        }
        __syncthreads();                       // all waves' chunk-c data visible

        // ---- 16x16 fp32 dot tile: 128 chained V_WMMA_F32_16X16X4_F32 ----
        v8f acc = {};
        const float* bptr = &ldsB[c & 1][nl * ROW_PAD + 2 * hi];
#pragma unroll
        for (int j = 0; j < D_STEPS; ++j) {
            v2f b = *(const v2f*)(bptr + 4 * j);
            acc = __builtin_amdgcn_wmma_f32_16x16x4_f32(
                false, afrag[j], false, b, (short)0, acc, false, false);
        }
        __syncthreads();   // done reading before this buffer is refilled (iter c+1 stages into buf c&1 ^ 1... buf[(c+2)&1])

        // ---- dist = cnorm[n] - 2*dot ; C layout: acc[v] is (M=v+8*hi, N=nl)
        float cn = cnorm[nBase + nl];
#pragma unroll
        for (int v = 0; v < 8; ++v) {
            float d = __builtin_fmaf(-2.0f, acc[v], cn);
            if (d < bestVal[v]) { bestVal[v] = d; bestIdx[v] = nBase + nl; }
        }
    }

    // ---- reduce (min,idx) across the 16 lanes of each N-group ----
#pragma unroll
    for (int v = 0; v < 8; ++v) {
        float bv = bestVal[v];
        int   bi = bestIdx[v];
#pragma unroll
        for (int m = 8; m >= 1; m >>= 1) {
            float ov = __shfl_xor(bv, m, 32);
            int   oi = __shfl_xor(bi, m, 32);
            if (ov < bv || (ov == bv && oi < bi)) { bv = ov; bi = oi; }
        }
        if (nl == 0) bestIdxSh[wave * 16 + v + 8 * hi] = bi;
    }
    __syncthreads();

    // ---- gather: z_q[row] = codebook[best[row]]; write indices ----
    {
        int r    = tid >> 1;          // 2 threads per row, 64 rows
        int half = tid & 1;
        int bi   = bestIdxSh[r];
        const float4* src = (const float4*)(cb + (size_t)bi * D_DIM) + half * 64;
        float4*       dst = (float4*)(zq + (size_t)(rowBase + r) * D_DIM) + half * 64;
#pragma unroll 8
        for (int i = 0; i < 64; ++i) dst[i] = src[i];
        if (half == 0) idx_out[rowBase + r] = bi;
    }
}

// --------------------------------------------------------------------------
extern "C" void kernel_launch(void* const* d_in, const int* in_sizes, int n_in,
                              void* d_out, int out_size, void* d_ws, size_t ws_size,
                              hipStream_t stream) {
    const float* z  = (const float*)d_in[0];   // [B, L, D] f32
    const float* cb = (const float*)d_in[1];   // [K, D]    f32

    const int BL = in_sizes[0] / D_DIM;        // 8192
    const int K  = in_sizes[1] / D_DIM;        // 8192

    float* zq      = (float*)d_out;            // first BL*D floats
    int*   idx_out = (int*)((float*)d_out + (size_t)BL * D_DIM);
    float* cnorm   = (float*)d_ws;             // K floats of scratch

    vq_cnorm_kernel<<<(K + 255) / 256, 256, 0, stream>>>(cb, cnorm, K);
    vq_argmin_kernel<<<BL / M_TILE, THREADS, 0, stream>>>(z, cb, cnorm, zq, idx_out, K);
}